// DGN_GraphProp_4406636446401
// MI455X (gfx1250) — compile-verified
//
#include <hip/hip_runtime.h>
#include <hip/hip_bf16.h>
#include <math.h>

#define D_IN   64
#define HID    128
#define NLAYERS 5
#define NGRAPHS 128
#define DOUT   10
#define HID3   (3 * HID)        // 384
#define HMID   (3 * HID / 2)    // 192

typedef __attribute__((ext_vector_type(2))) float v2f;
typedef __attribute__((ext_vector_type(8))) float v8f;

// -------------------------------------------------------------------------
// Persistent WMMA fp32 GEMM:  C[nrows x 128] = A[nrows x K] * W[K x 128] (+bias)
// Block = 256 threads = 8 waves. Each block stages W into LDS ONCE, then
// strides over 16-row tiles (amortizes the 64KB LDS staging ~4x vs one
// tile per block). Wave w owns the 16x16 tile at columns [16w, 16w+16).
//
// Fragment layouts (wave32):
//  A 16x4 f32 (ISA table): lane L<16: {A[L][k], A[L][k+1]}
//                          lane L>=16: {A[L-16][k+2], A[L-16][k+3]}
//  B 4x16 f32 (C/D-style row striping): vgpr r, lanes 0-15 -> row k+r,
//                                       lanes 16-31 -> row k+r+2
//  C/D 16x16 f32: vgpr r, lanes 0-15 -> M=r, lanes 16-31 -> M=r+8
// -------------------------------------------------------------------------
__global__ __launch_bounds__(256)
void gemm_wmma_f32(const float* __restrict__ A, const float* __restrict__ W,
                   const float* __restrict__ bias, float* __restrict__ C,
                   int nrows, int K, int ntiles)
{
    extern __shared__ float lw[];
    const int tid = threadIdx.x;

    // cooperative LDS staging of W (K*128 floats, float4 chunks) -- once
    const int nf4 = (K * HID) >> 2;
    const float4* Wv = (const float4*)W;
    float4* lwv = (float4*)lw;
    for (int i = tid; i < nf4; i += 256) lwv[i] = Wv[i];
    __syncthreads();

    const int wave = tid >> 5;
    const int lane = tid & 31;
    const int col0 = wave * 16;
    const int lrow = lane & 15;
    const bool hi  = lane >= 16;
    const float bb = bias ? bias[col0 + lrow] : 0.0f;

    for (int tile = blockIdx.x; tile < ntiles; tile += gridDim.x) {
        const int row0 = tile * 16;
        int arow = row0 + lrow;
        if (arow >= nrows) arow = nrows - 1;        // clamp; store is guarded
        const float* Arow = A + (size_t)arow * (size_t)K;

        // prefetch next tile's A rows (global_prefetch_b8) to overlap with WMMA
        const int ntile = tile + gridDim.x;
        if (ntile < ntiles) {
            int prow = ntile * 16 + lrow;
            if (prow >= nrows) prow = nrows - 1;
            const float* P = A + (size_t)prow * (size_t)K;
            for (int p = 0; p < K; p += 32) __builtin_prefetch(P + p, 0, 0);
        }

        v8f acc = {};
        for (int k = 0; k < K; k += 4) {
            float2 a2 = *(const float2*)(Arow + k + (hi ? 2 : 0));
            v2f a; a.x = a2.x; a.y = a2.y;
            const int brow = k + (hi ? 2 : 0);
            v2f b;
            b.x = lw[(brow + 0) * HID + col0 + lrow];
            b.y = lw[(brow + 1) * HID + col0 + lrow];
            acc = __builtin_amdgcn_wmma_f32_16x16x4_f32(
                      false, a, false, b, (short)0, acc, false, false);
        }

        const int crow_base = row0 + (hi ? 8 : 0);
        const int ccol = col0 + lrow;
#pragma unroll
        for (int r = 0; r < 8; ++r) {
            int cr = crow_base + r;
            if (cr < nrows) C[(size_t)cr * HID + ccol] = acc[r] + bb;
        }
    }
}

// ------------------------- degree / norm precompute -----------------------
__global__ void k_init_deg(float* deg, int N) {
    int i = blockIdx.x * blockDim.x + threadIdx.x;
    if (i < N) deg[i] = 1.0f;                      // self-loop contribution
}
__global__ void k_accum_deg(const int* __restrict__ dst, float* deg, int E) {
    int e = blockIdx.x * blockDim.x + threadIdx.x;
    if (e < E) atomicAdd(&deg[dst[e]], 1.0f);
}
__global__ void k_make_dinv(float* deg, int N) {   // deg >= 1 always
    int i = blockIdx.x * blockDim.x + threadIdx.x;
    if (i < N) deg[i] = rsqrtf(deg[i]);
}
__global__ void k_make_norm(const int* __restrict__ src, const int* __restrict__ dst,
                            const float* __restrict__ dinv, float* norm, int E) {
    int e = blockIdx.x * blockDim.x + threadIdx.x;
    if (e < E) norm[e] = dinv[src[e]] * dinv[dst[e]];
}

// self-loop message initializes agg (no atomics): agg[i] = hw[i] * dinv[i]^2
__global__ void k_self_init(const float* __restrict__ hw, const float* __restrict__ dinv,
                            float* __restrict__ agg, int N) {
    int t = blockIdx.x * blockDim.x + threadIdx.x;
    int node = t >> 5;
    if (node >= N) return;
    int c = (t & 31) * 4;
    float s = dinv[node]; s = s * s;
    float4 v = *(const float4*)(hw + (size_t)node * HID + c);
    float4 o; o.x = v.x * s; o.y = v.y * s; o.z = v.z * s; o.w = v.w * s;
    *(float4*)(agg + (size_t)node * HID + c) = o;
}

// one wave32 per edge; lane handles 4 channels; fp32 atomics land in L2
// (hw and agg are each 51.2 MB -> both L2-resident on the 192 MB L2)
__global__ __launch_bounds__(256)
void k_edge_scatter(const int* __restrict__ src, const int* __restrict__ dst,
                    const float* __restrict__ norm, const float* __restrict__ hw,
                    float* __restrict__ agg, int E) {
    int e = blockIdx.x * 8 + (threadIdx.x >> 5);
    if (e >= E) return;
    int lane = threadIdx.x & 31;
    int s = src[e], d = dst[e];
    float w = norm[e];
    float4 v = *(const float4*)(hw + (size_t)s * HID + lane * 4);
    float* ap = agg + (size_t)d * HID + lane * 4;
    atomicAdd(ap + 0, v.x * w);
    atomicAdd(ap + 1, v.y * w);
    atomicAdd(ap + 2, v.z * w);
    atomicAdd(ap + 3, v.w * w);
}

__global__ void k_bias_tanh(const float* __restrict__ agg, const float* __restrict__ b,
                            float* __restrict__ h, int n) {
    int i = blockIdx.x * blockDim.x + threadIdx.x;
    if (i < n) h[i] = tanhf(agg[i] + b[i & (HID - 1)]);
}

// ------------------------------- pooling ----------------------------------
__global__ void k_pool_init(float* sum, float* mx, float* cnt, int n) {
    int i = blockIdx.x * blockDim.x + threadIdx.x;
    if (i < n) { sum[i] = 0.0f; mx[i] = -INFINITY; }
    if (i < NGRAPHS) cnt[i] = 0.0f;
}
__device__ __forceinline__ void atomicMaxF(float* addr, float val) {
    unsigned int* ua = (unsigned int*)addr;
    unsigned int old = *ua;
    while (__uint_as_float(old) < val) {
        unsigned int assumed = old;
        old = atomicCAS(ua, assumed, __float_as_uint(val));
        if (old == assumed) break;
    }
}
__global__ void k_pool_accum(const float* __restrict__ h, const int* __restrict__ batch,
                             float* sum, float* mx, int N) {
    int t = blockIdx.x * blockDim.x + threadIdx.x;
    int node = t >> 7;           // /HID
    if (node >= N) return;
    int c = t & (HID - 1);
    int g = batch[node];
    float v = h[t];
    atomicAdd(&sum[g * HID + c], v);
    atomicMaxF(&mx[g * HID + c], v);
}
__global__ void k_cnt_accum(const int* __restrict__ batch, float* cnt, int N) {
    int i = blockIdx.x * blockDim.x + threadIdx.x;
    if (i < N) atomicAdd(&cnt[batch[i]], 1.0f);
}
__global__ void k_pool_final(const float* __restrict__ sum, const float* __restrict__ mx,
                             const float* __restrict__ cnt, float* __restrict__ g) {
    int t = blockIdx.x * blockDim.x + threadIdx.x;
    if (t >= NGRAPHS * HID) return;
    int gi = t / HID, c = t % HID;
    float s = sum[t];
    float m = mx[t];
    float n = cnt[gi];
    float* row = g + (size_t)gi * HID3;
    row[c]           = s;
    row[HID + c]     = isfinite(m) ? m : 0.0f;
    row[2 * HID + c] = s / fmaxf(n, 1.0f);
}

// ------------------------------- MLP tail ---------------------------------
__global__ void k_mlp1(const float* __restrict__ g, const float* __restrict__ W1,
                       const float* __restrict__ b1, float* __restrict__ o) {
    int t = blockIdx.x * blockDim.x + threadIdx.x;
    if (t >= NGRAPHS * HMID) return;
    int gi = t / HMID, j = t % HMID;
    const float* gr = g + (size_t)gi * HID3;
    float acc = b1[j];
    for (int k = 0; k < HID3; ++k) acc = fmaf(gr[k], W1[k * HMID + j], acc);
    o[t] = acc > 0.0f ? acc : 0.01f * acc;
}
__global__ void k_mlp2(const float* __restrict__ o, const float* __restrict__ W2,
                       const float* __restrict__ b2, float* __restrict__ out) {
    int t = blockIdx.x * blockDim.x + threadIdx.x;
    if (t >= NGRAPHS * DOUT) return;
    int gi = t / DOUT, j = t % DOUT;
    const float* orow = o + (size_t)gi * HMID;
    float acc = b2[j];
    for (int k = 0; k < HMID; ++k) acc = fmaf(orow[k], W2[k * DOUT + j], acc);
    out[t] = acc > 0.0f ? acc : 0.01f * acc;
}

// --------------------------------------------------------------------------
static inline size_t align256(size_t x) { return (x + 255) & ~(size_t)255; }

extern "C" void kernel_launch(void* const* d_in, const int* in_sizes, int n_in,
                              void* d_out, int out_size, void* d_ws, size_t ws_size,
                              hipStream_t stream) {
    const float* x       = (const float*)d_in[0];
    const int*   eidx    = (const int*)d_in[1];
    const int*   batch   = (const int*)d_in[2];
    const float* W_emb   = (const float*)d_in[3];
    const float* b_emb   = (const float*)d_in[4];
    const float* W_conv  = (const float*)d_in[5];
    const float* b_conv  = (const float*)d_in[6];
    const float* W1      = (const float*)d_in[7];
    const float* b1      = (const float*)d_in[8];
    const float* W2      = (const float*)d_in[9];
    const float* b2      = (const float*)d_in[10];

    const int N = in_sizes[0] / D_IN;
    const int E = in_sizes[1] / 2;
    const int* src = eidx;
    const int* dst = eidx + E;

    // workspace carve-out
    char* base = (char*)d_ws;
    size_t off = 0;
    auto carve = [&](size_t bytes) -> void* {
        void* p = base + off; off = align256(off + bytes); return p;
    };
    float* h    = (float*)carve((size_t)N * HID * 4);
    float* hw   = (float*)carve((size_t)N * HID * 4);
    float* agg  = (float*)carve((size_t)N * HID * 4);
    float* dinv = (float*)carve((size_t)N * 4);          // deg -> dinv in place
    float* norm = (float*)carve((size_t)E * 4);
    float* sum  = (float*)carve((size_t)NGRAPHS * HID * 4);
    float* mx   = (float*)carve((size_t)NGRAPHS * HID * 4);
    float* cnt  = (float*)carve((size_t)NGRAPHS * 4);
    float* gcat = (float*)carve((size_t)NGRAPHS * HID3 * 4);
    float* mo   = (float*)carve((size_t)NGRAPHS * HMID * 4);

    const int T = 256;
    auto cdiv = [](long long a, long long b) { return (int)((a + b - 1) / b); };

    const int ntiles = cdiv(N, 16);
    const int gemm_blocks = ntiles < 1536 ? ntiles : 1536;  // persistent grid

    // degree / normalization
    k_init_deg<<<cdiv(N, T), T, 0, stream>>>(dinv, N);
    k_accum_deg<<<cdiv(E, T), T, 0, stream>>>(dst, dinv, E);
    k_make_dinv<<<cdiv(N, T), T, 0, stream>>>(dinv, N);
    k_make_norm<<<cdiv(E, T), T, 0, stream>>>(src, dst, dinv, norm, E);

    // embedding: h = x @ W_emb + b_emb  (K = 64)
    gemm_wmma_f32<<<gemm_blocks, T, D_IN * HID * 4, stream>>>(x, W_emb, b_emb, h, N, D_IN, ntiles);

    // 5 GCN layers
    for (int l = 0; l < NLAYERS; ++l) {
        const float* Wl = W_conv + (size_t)l * HID * HID;
        const float* bl = b_conv + (size_t)l * HID;
        gemm_wmma_f32<<<gemm_blocks, T, HID * HID * 4, stream>>>(h, Wl, nullptr, hw, N, HID, ntiles);
        k_self_init<<<cdiv((long long)N * 32, T), T, 0, stream>>>(hw, dinv, agg, N);
        k_edge_scatter<<<cdiv(E, 8), T, 0, stream>>>(src, dst, norm, hw, agg, E);
        k_bias_tanh<<<cdiv((long long)N * HID, T), T, 0, stream>>>(agg, bl, h, N * HID);
    }

    // pooling
    k_pool_init<<<cdiv(NGRAPHS * HID, T), T, 0, stream>>>(sum, mx, cnt, NGRAPHS * HID);
    k_pool_accum<<<cdiv((long long)N * HID, T), T, 0, stream>>>(h, batch, sum, mx, N);
    k_cnt_accum<<<cdiv(N, T), T, 0, stream>>>(batch, cnt, N);
    k_pool_final<<<cdiv(NGRAPHS * HID, T), T, 0, stream>>>(sum, mx, cnt, gcat);

    // MLP head
    k_mlp1<<<cdiv(NGRAPHS * HMID, T), T, 0, stream>>>(gcat, W1, b1, mo);
    k_mlp2<<<cdiv(NGRAPHS * DOUT, T), T, 0, stream>>>(mo, W2, b2, (float*)d_out);
}